// sLSTMBlock_86045374808605
// MI455X (gfx1250) — compile-verified
//
#include <hip/hip_runtime.h>
#include <math.h>
#include <stdint.h>

typedef __attribute__((ext_vector_type(2))) float v2f;
typedef __attribute__((ext_vector_type(8))) float v8f;

#define B_   4096
#define D_   2048
#define HID_ 2048
#define NH_  8
#define HD_  256
#define UP_  2730
#define EPS_ 1e-3f

__device__ __forceinline__ v8f wmma_f32(v2f a, v2f b, v8f c) {
  // D = A(16x4) * B(4x16) + C, fp32 on the matrix pipe
  return __builtin_amdgcn_wmma_f32_16x16x4_f32(false, a, false, b, (short)0, c, false, false);
}

// ---- CDNA5 async global->LDS staging (ASYNCcnt-tracked, see CDNA5 ISA) --
__device__ __forceinline__ void async_b128(void* lds, const void* g) {
  asm volatile("global_load_async_to_lds_b128 %0, %1, off"
               :: "v"((uint32_t)(uintptr_t)lds),
                  "v"((unsigned long long)(uintptr_t)g) : "memory");
}
__device__ __forceinline__ void async_b32(void* lds, const void* g) {
  asm volatile("global_load_async_to_lds_b32 %0, %1, off"
               :: "v"((uint32_t)(uintptr_t)lds),
                  "v"((unsigned long long)(uintptr_t)g) : "memory");
}
__device__ __forceinline__ void wait_async() {
  asm volatile("s_wait_asynccnt 0x0" ::: "memory");
}

// ---------------- Kernel 0: LayerNorm + causal-conv scalar + SiLU ----------------
__global__ __launch_bounds__(256) void k_lnconv(const float* __restrict__ x,
    const float* __restrict__ lng, const float* __restrict__ lnb,
    const float* __restrict__ ck, const float* __restrict__ cb,
    float* __restrict__ xc)
{
  __shared__ float red[256];
  const int b = blockIdx.x, t = threadIdx.x;
  const float* xr = x + (size_t)b * D_;
  float s = 0.f, s2 = 0.f;
  float xv[8];
  for (int j = 0; j < 8; ++j) { float v = xr[t + 256*j]; xv[j] = v; s += v; s2 += v*v; }
  red[t] = s; __syncthreads();
  for (int off = 128; off > 0; off >>= 1) { if (t < off) red[t] += red[t+off]; __syncthreads(); }
  const float mean = red[0] / (float)D_;
  __syncthreads();
  red[t] = s2; __syncthreads();
  for (int off = 128; off > 0; off >>= 1) { if (t < off) red[t] += red[t+off]; __syncthreads(); }
  const float var = red[0] / (float)D_ - mean*mean;
  __syncthreads();
  const float rstd = rsqrtf(var + EPS_);
  float d = 0.f;
  for (int j = 0; j < 8; ++j) {
    int c = t + 256*j;
    float xn = (xv[j] - mean) * rstd * lng[c] + lnb[c];
    d += xn * ck[3*D_ + c];           // only the last causal-conv tap is live at T=1
  }
  red[t] = d; __syncthreads();
  for (int off = 128; off > 0; off >>= 1) { if (t < off) red[t] += red[t+off]; __syncthreads(); }
  if (t == 0) { float sc = red[0] + cb[0]; xc[b] = sc / (1.f + expf(-sc)); }
}

// ---------------- Kernel 0b: column sums of Wi_w / Wf_w (rank-1 x_conv path) ------
__global__ __launch_bounds__(256) void k_colsum(const float* __restrict__ Wi,
    const float* __restrict__ Wf, float* __restrict__ csI, float* __restrict__ csF)
{
  const int idx = blockIdx.x*256 + threadIdx.x;   // 0..2047
  const int h = idx >> 8, o = idx & 255;
  const float* wi = Wi + (size_t)h*HD_*HD_ + o;
  const float* wf = Wf + (size_t)h*HD_*HD_ + o;
  float si = 0.f, sf = 0.f;
  for (int i = 0; i < HD_; ++i) { si += wi[(size_t)i*HD_]; sf += wf[(size_t)i*HD_]; }
  csI[idx] = si; csF[idx] = sf;
}

// ------- Kernel 1: block-diag gates (WMMA, M=64/wave-strip) + cell + GroupNorm ----
__global__ __launch_bounds__(512) void k_gates(
    const float* __restrict__ x,  const float* __restrict__ hp,
    const float* __restrict__ cp, const float* __restrict__ np_,
    const float* __restrict__ mp,
    const float* __restrict__ Wz, const float* __restrict__ Wzb,
    const float* __restrict__ Wib, const float* __restrict__ Wfb,
    const float* __restrict__ Wo, const float* __restrict__ Wob,
    const float* __restrict__ Rz, const float* __restrict__ Rzb,
    const float* __restrict__ Ri, const float* __restrict__ Rib,
    const float* __restrict__ Rf, const float* __restrict__ Rfb,
    const float* __restrict__ Ro, const float* __restrict__ Rob,
    const float* __restrict__ gng, const float* __restrict__ gnb,
    const float* __restrict__ xc, const float* __restrict__ csI,
    const float* __restrict__ csF,
    float* __restrict__ h_out, float* __restrict__ c_out,
    float* __restrict__ n_out, float* __restrict__ m_out,
    float* __restrict__ onorm)
{
  __shared__ __align__(16) float xs[64][260];  // stride 260: 16B rows + bank-walked cols
  __shared__ __align__(16) float hs[64][260];
  __shared__ float ht[64][257];
  const int tid   = threadIdx.x;
  const int head  = blockIdx.y;
  const int mbase = blockIdx.x * 64;

  // async-stage x and h head-slices: 64 rows x 256 cols, 16B segments
  for (int j = 0; j < 8; ++j) {
    int idx = tid + j*512;                      // 4096 segments per matrix
    int r = idx >> 6, c4 = (idx & 63) * 4;
    size_t g = (size_t)(mbase + r)*D_ + head*HD_ + c4;
    async_b128(&xs[r][c4], x + g);
    async_b128(&hs[r][c4], hp + g);
  }
  wait_async();
  __syncthreads();

  const int wid = tid >> 5, lane = tid & 31;
  const int nn = lane & 15, kh = lane >> 4;
  const int col = wid*16 + nn;                  // wave owns one 16-col strip
  const size_t wb = (size_t)head * HD_ * HD_;
  const float *wz = Wz + wb, *wo = Wo + wb;
  const float *rz = Rz + wb, *ro = Ro + wb, *ri = Ri + wb, *rf = Rf + wb;

  v8f aZ[4] = {}, aO[4] = {}, aI[4] = {}, aF[4] = {};
  for (int k0 = 0; k0 < HD_; k0 += 4) {         // A = x head-slice, gates z/o
    int ka = k0 + 2*kh;
    v2f a[4];
#pragma unroll
    for (int mi = 0; mi < 4; ++mi) { a[mi].x = xs[mi*16+nn][ka]; a[mi].y = xs[mi*16+nn][ka+1]; }
    v2f b;
    b.x = wz[(size_t)ka*HD_ + col]; b.y = wz[(size_t)(ka+1)*HD_ + col];
#pragma unroll
    for (int mi = 0; mi < 4; ++mi) aZ[mi] = wmma_f32(a[mi], b, aZ[mi]);
    b.x = wo[(size_t)ka*HD_ + col]; b.y = wo[(size_t)(ka+1)*HD_ + col];
#pragma unroll
    for (int mi = 0; mi < 4; ++mi) aO[mi] = wmma_f32(a[mi], b, aO[mi]);
  }
  for (int k0 = 0; k0 < HD_; k0 += 4) {         // A = h_prev head-slice, gates z/o/i/f
    int ka = k0 + 2*kh;
    v2f a[4];
#pragma unroll
    for (int mi = 0; mi < 4; ++mi) { a[mi].x = hs[mi*16+nn][ka]; a[mi].y = hs[mi*16+nn][ka+1]; }
    v2f b;
    b.x = rz[(size_t)ka*HD_ + col]; b.y = rz[(size_t)(ka+1)*HD_ + col];
#pragma unroll
    for (int mi = 0; mi < 4; ++mi) aZ[mi] = wmma_f32(a[mi], b, aZ[mi]);
    b.x = ro[(size_t)ka*HD_ + col]; b.y = ro[(size_t)(ka+1)*HD_ + col];
#pragma unroll
    for (int mi = 0; mi < 4; ++mi) aO[mi] = wmma_f32(a[mi], b, aO[mi]);
    b.x = ri[(size_t)ka*HD_ + col]; b.y = ri[(size_t)(ka+1)*HD_ + col];
#pragma unroll
    for (int mi = 0; mi < 4; ++mi) aI[mi] = wmma_f32(a[mi], b, aI[mi]);
    b.x = rf[(size_t)ka*HD_ + col]; b.y = rf[(size_t)(ka+1)*HD_ + col];
#pragma unroll
    for (int mi = 0; mi < 4; ++mi) aF[mi] = wmma_f32(a[mi], b, aF[mi]);
  }

  const int gcol = head*HD_ + col;
  const float bz = Wzb[gcol] + Rzb[gcol];
  const float bo = Wob[gcol] + Rob[gcol];
  const float bi = Wib[gcol] + Rib[gcol];
  const float bf = Wfb[gcol] + Rfb[gcol];
  const float cI = csI[gcol], cF = csF[gcol];
#pragma unroll
  for (int mi = 0; mi < 4; ++mi) {
#pragma unroll
    for (int v = 0; v < 8; ++v) {
      int mr = mi*16 + v + 8*kh;
      int brow = mbase + mr;
      size_t idx = (size_t)brow * HID_ + gcol;
      float xcb = xc[brow];
      float zt = tanhf(aZ[mi][v] + bz);
      float ot = 1.f / (1.f + expf(-(aO[mi][v] + bo)));
      float it = aI[mi][v] + xcb*cI + bi;
      float ft = aF[mi][v] + xcb*cF + bf;
      float mpv = mp[idx];
      float mt = fmaxf(ft + mpv, it);
      float ig = expf(it - mt);
      float fg = expf(ft + mpv - mt);
      float ct = fg*cp[idx] + ig*zt;
      float nt = fg*np_[idx] + ig;
      float hv = ot*ct/nt;
      h_out[idx] = hv; c_out[idx] = ct; n_out[idx] = nt; m_out[idx] = mt;
      ht[mr][col] = hv;
    }
  }
  __syncthreads();

  // GroupNorm: head's 256 cols == one group; each wave32 handles 4 rows
  for (int q = 0; q < 4; ++q) {
    const int row = wid + 16*q;
    float s = 0.f, s2 = 0.f;
    float vals[8];
    for (int j = 0; j < 8; ++j) { float v = ht[row][lane + 32*j]; vals[j] = v; s += v; s2 += v*v; }
    for (int off = 16; off > 0; off >>= 1) { s += __shfl_down(s, off, 32); s2 += __shfl_down(s2, off, 32); }
    s = __shfl(s, 0, 32); s2 = __shfl(s2, 0, 32);
    const float mean = s / (float)HD_;
    const float var  = s2 / (float)HD_ - mean*mean;
    const float rstd = rsqrtf(var + EPS_);
    const int brow = mbase + row;
    for (int j = 0; j < 8; ++j) {
      int c = lane + 32*j;
      int gc = head*HD_ + c;
      onorm[(size_t)brow*HID_ + gc] = (vals[j] - mean)*rstd*gng[gc] + gnb[gc];
    }
  }
}

// ---------------- Kernel 2: upL/upR GEMM (BM=64,BN=128) + exact-GELU gating -------
__global__ __launch_bounds__(256) void k_up(
    const float* __restrict__ A, const float* __restrict__ BL,
    const float* __restrict__ BR, const float* __restrict__ bL,
    const float* __restrict__ bR, float* __restrict__ gated)
{
  __shared__ __align__(16) float As[64][68];
  __shared__ __align__(16) float Bls[64][128];
  __shared__ __align__(16) float Brs[64][128];
  const int tid = threadIdx.x;
  const int mb  = blockIdx.x * 64;
  const int nb0 = blockIdx.y * 128;
  const int wid = tid >> 5, lane = tid & 31;
  const int nn = lane & 15, kh = lane >> 4;
  const int wm = (wid >> 2) * 32;               // 0 / 32
  const int wn = (wid & 3) * 32;                // 0 / 32 / 64 / 96
  v8f aL[2][2] = {}, aR[2][2] = {};             // [mi][ni]
  for (int kc = 0; kc < HID_; kc += 64) {
    if (kc + 64 < HID_) __builtin_prefetch(BL + (size_t)(kc + 64)*UP_ + nb0, 0, 1);
    for (int j = 0; j < 4; ++j) {               // A: 64x64, b128 (stride 2048 aligned)
      int idx = tid + j*256; int r = idx >> 4, c4 = (idx & 15) * 4;
      async_b128(&As[r][c4], A + (size_t)(mb + r)*HID_ + kc + c4);
    }
    for (int j = 0; j < 32; ++j) {              // B: 64x128 x2, b32 (stride 2730)
      int idx = tid + j*256; int r = idx >> 7, c = idx & 127;
      int n = nb0 + c;
      if (n < UP_) {
        size_t g = (size_t)(kc + r)*UP_ + n;
        async_b32(&Bls[r][c], BL + g);
        async_b32(&Brs[r][c], BR + g);
      }
    }
    wait_async();
    __syncthreads();
    for (int k0 = 0; k0 < 64; k0 += 4) {
      int ka = k0 + 2*kh;
      v2f a0, a1;
      a0.x = As[wm+nn][ka];    a0.y = As[wm+nn][ka+1];
      a1.x = As[wm+16+nn][ka]; a1.y = As[wm+16+nn][ka+1];
      v2f b;
      b.x = Bls[ka][wn+nn];    b.y = Bls[ka+1][wn+nn];
      aL[0][0] = wmma_f32(a0, b, aL[0][0]); aL[1][0] = wmma_f32(a1, b, aL[1][0]);
      b.x = Bls[ka][wn+16+nn]; b.y = Bls[ka+1][wn+16+nn];
      aL[0][1] = wmma_f32(a0, b, aL[0][1]); aL[1][1] = wmma_f32(a1, b, aL[1][1]);
      b.x = Brs[ka][wn+nn];    b.y = Brs[ka+1][wn+nn];
      aR[0][0] = wmma_f32(a0, b, aR[0][0]); aR[1][0] = wmma_f32(a1, b, aR[1][0]);
      b.x = Brs[ka][wn+16+nn]; b.y = Brs[ka+1][wn+16+nn];
      aR[0][1] = wmma_f32(a0, b, aR[0][1]); aR[1][1] = wmma_f32(a1, b, aR[1][1]);
    }
    __syncthreads();
  }
#pragma unroll
  for (int ni = 0; ni < 2; ++ni) {
    int n = nb0 + wn + ni*16 + nn;
    if (n >= UP_) continue;
    float lb = bL[n], rb = bR[n];
#pragma unroll
    for (int mi = 0; mi < 2; ++mi)
#pragma unroll
      for (int v = 0; v < 8; ++v) {
        int mr = wm + mi*16 + v + 8*kh;
        float L = aL[mi][ni][v] + lb;
        float R = aR[mi][ni][v] + rb;
        float g = 0.5f * R * (1.f + erff(R * 0.70710678118654752f)); // exact GELU
        gated[(size_t)(mb + mr)*UP_ + n] = L * g;
      }
  }
}

// ---------------- Kernel 3: down GEMM (BM=128,BN=128) + bias + residual ------------
__global__ __launch_bounds__(256) void k_down(
    const float* __restrict__ G, const float* __restrict__ Dw,
    const float* __restrict__ Db, const float* __restrict__ x,
    float* __restrict__ out)
{
  __shared__ __align__(16) float As[128][68];
  __shared__ __align__(16) float Bs[64][128];
  const int tid = threadIdx.x;
  const int mb  = blockIdx.x * 128;
  const int nb0 = blockIdx.y * 128;
  const int wid = tid >> 5, lane = tid & 31;
  const int nn = lane & 15, kh = lane >> 4;
  const int wm = (wid >> 1) * 32;               // 0/32/64/96
  const int wn = (wid & 1) * 64;                // 0/64
  v8f acc[2][4] = {};                           // [mi][ni]
  for (int kc = 0; kc < UP_; kc += 64) {        // K=2730, zero-filled tail
    for (int j = 0; j < 32; ++j) {              // A: 128x64, b32 (stride 2730)
      int idx = tid + j*256; int r = idx >> 6, c = idx & 63;
      int k = kc + c;
      if (k < UP_) async_b32(&As[r][c], G + (size_t)(mb + r)*UP_ + k);
      else         As[r][c] = 0.f;
    }
    for (int j = 0; j < 8; ++j) {               // B: 64x128, b128 (stride 2048 aligned)
      int idx = tid + j*256; int r = idx >> 5, c4 = (idx & 31) * 4;
      int k = kc + r;
      if (k < UP_) async_b128(&Bs[r][c4], Dw + (size_t)k*D_ + nb0 + c4);
      else { Bs[r][c4] = 0.f; Bs[r][c4+1] = 0.f; Bs[r][c4+2] = 0.f; Bs[r][c4+3] = 0.f; }
    }
    wait_async();
    __syncthreads();
    for (int k0 = 0; k0 < 64; k0 += 4) {
      int ka = k0 + 2*kh;
      v2f a0, a1;
      a0.x = As[wm+nn][ka];    a0.y = As[wm+nn][ka+1];
      a1.x = As[wm+16+nn][ka]; a1.y = As[wm+16+nn][ka+1];
#pragma unroll
      for (int ni = 0; ni < 4; ++ni) {
        v2f b;
        b.x = Bs[ka][wn + ni*16 + nn]; b.y = Bs[ka+1][wn + ni*16 + nn];
        acc[0][ni] = wmma_f32(a0, b, acc[0][ni]);
        acc[1][ni] = wmma_f32(a1, b, acc[1][ni]);
      }
    }
    __syncthreads();
  }
#pragma unroll
  for (int ni = 0; ni < 4; ++ni) {
    int n = nb0 + wn + ni*16 + nn;
    float db = Db[n];
#pragma unroll
    for (int mi = 0; mi < 2; ++mi)
#pragma unroll
      for (int v = 0; v < 8; ++v) {
        int mr = wm + mi*16 + v + 8*kh;
        size_t idx = (size_t)(mb + mr)*D_ + n;
        out[idx] = acc[mi][ni][v] + db + x[idx];
      }
  }
}

extern "C" void kernel_launch(void* const* d_in, const int* in_sizes, int n_in,
                              void* d_out, int out_size, void* d_ws, size_t ws_size,
                              hipStream_t stream) {
  (void)in_sizes; (void)n_in; (void)out_size; (void)ws_size;
  const float* x    = (const float*)d_in[0];
  const float* hp   = (const float*)d_in[1];
  const float* cp   = (const float*)d_in[2];
  const float* np_  = (const float*)d_in[3];
  const float* mp   = (const float*)d_in[4];
  const float* lng  = (const float*)d_in[5];
  const float* lnb  = (const float*)d_in[6];
  const float* ck   = (const float*)d_in[7];
  const float* cb   = (const float*)d_in[8];
  const float* Wz   = (const float*)d_in[9];
  const float* Wzb  = (const float*)d_in[10];
  const float* Wi   = (const float*)d_in[11];
  const float* Wib  = (const float*)d_in[12];
  const float* Wf   = (const float*)d_in[13];
  const float* Wfb  = (const float*)d_in[14];
  const float* Wo   = (const float*)d_in[15];
  const float* Wob  = (const float*)d_in[16];
  const float* Rz   = (const float*)d_in[17];
  const float* Rzb  = (const float*)d_in[18];
  const float* Ri   = (const float*)d_in[19];
  const float* Rib  = (const float*)d_in[20];
  const float* Rf   = (const float*)d_in[21];
  const float* Rfb  = (const float*)d_in[22];
  const float* Ro   = (const float*)d_in[23];
  const float* Rob  = (const float*)d_in[24];
  const float* gng  = (const float*)d_in[25];
  const float* gnb  = (const float*)d_in[26];
  const float* uLw  = (const float*)d_in[27];
  const float* uLb  = (const float*)d_in[28];
  const float* uRw  = (const float*)d_in[29];
  const float* uRb  = (const float*)d_in[30];
  const float* dW   = (const float*)d_in[31];
  const float* dB   = (const float*)d_in[32];

  float* out = (float*)d_out;
  const size_t BD = (size_t)B_ * D_;
  float* f_out = out;
  float* h_out = out + BD;
  float* c_out = out + 2*BD;
  float* n_out = out + 3*BD;
  float* m_out = out + 4*BD;

  float* ws    = (float*)d_ws;
  float* xc    = ws;                 // [4096]
  float* csI   = ws + 4096;          // [2048]
  float* csF   = ws + 6144;          // [2048]
  float* onorm = ws + 8192;          // [B*HID]
  float* gated = onorm + BD;         // [B*UP]

  k_lnconv<<<B_, 256, 0, stream>>>(x, lng, lnb, ck, cb, xc);
  k_colsum<<<8, 256, 0, stream>>>(Wi, Wf, csI, csF);
  k_gates<<<dim3(B_/64, NH_), 512, 0, stream>>>(x, hp, cp, np_, mp,
      Wz, Wzb, Wib, Wfb, Wo, Wob, Rz, Rzb, Ri, Rib, Rf, Rfb, Ro, Rob,
      gng, gnb, xc, csI, csF, h_out, c_out, n_out, m_out, onorm);
  k_up<<<dim3(B_/64, (UP_ + 127)/128), 256, 0, stream>>>(onorm, uLw, uRw, uLb, uRb, gated);
  k_down<<<dim3(B_/128, D_/128), 256, 0, stream>>>(gated, dW, dB, x, f_out);
}